// Layer_1382979469614
// MI455X (gfx1250) — compile-verified
//
#include <hip/hip_runtime.h>

// MI455X / gfx1250, wave32. fp32 WMMA path: V_WMMA_F32_16X16X4_F32.
// Async staging: GLOBAL_LOAD_ASYNC_TO_LDS_B128 (ASYNCcnt) via inline asm.

typedef float v2f __attribute__((ext_vector_type(2)));
typedef float v8f __attribute__((ext_vector_type(8)));

#define BATCH 4
#define SEQ   2048
#define DIM   256
#define LDP   260   // padded LDS row stride; row pitch 1040 B (16B aligned), 260%64=4

static __device__ __forceinline__ v8f wmma_f32_16x16x4(v2f a, v2f b, v8f c) {
  // 8 args: (neg_a, A, neg_b, B, c_mod, C, reuse_a, reuse_b)
  return __builtin_amdgcn_wmma_f32_16x16x4_f32(false, a, false, b, (short)0, c,
                                               false, false);
}

// One 16-byte async copy global -> LDS. vdst = LDS byte address (generic-ptr
// low 32 bits == wave-relative LDS offset per the flat-aperture rule).
static __device__ __forceinline__ void async_load_b128(unsigned lds_off,
                                                       const void* gaddr) {
  asm volatile("global_load_async_to_lds_b128 %0, %1, off"
               :: "v"(lds_off), "v"(gaddr)
               : "memory");
}

static __device__ __forceinline__ void wait_async0() {
  asm volatile("s_wait_asynccnt 0x0" ::: "memory");
}

// ---------------------------------------------------------------------------
// Kernel 1: h = x @ M^T, then LayerNorm -> xt (workspace).
// Grid: (B*S)/16 = 512 blocks. Block: 256 threads = 8 waves.
// Each block computes 16 rows x 256 cols; each wave owns two 16x16 N-tiles.
// ---------------------------------------------------------------------------
__global__ __launch_bounds__(256) void k_gemm_ln(
    const float* __restrict__ x, const float* __restrict__ M,
    const float* __restrict__ gamma, const float* __restrict__ beta,
    float* __restrict__ xt) {
  __shared__ float lds[16 * LDP];

  const int tid  = threadIdx.x;
  const int wave = tid >> 5;
  const int lane = tid & 31;
  const size_t R0 = (size_t)blockIdx.x * 16;   // first flat row (b*S+s)

  __builtin_prefetch(M, 0, 0);  // global_prefetch_b8

  // Async-stage 16x256 x-tile into LDS: 1024 float4 transfers, 4 per thread.
  for (int i = tid; i < 16 * 64; i += 256) {
    const int r = i >> 6, c4 = i & 63;
    const unsigned loff = (unsigned)(size_t)&lds[r * LDP + c4 * 4];
    async_load_b128(loff, x + (R0 + r) * DIM + c4 * 4);
  }
  wait_async0();
  __syncthreads();

  const int mrow  = lane & 15;           // M index (and N index for B frags)
  const int khalf = (lane >> 4) << 1;    // lanes 0-15 -> K{0,1}; 16-31 -> K{2,3}
  const int n0 = wave * 32 + mrow;       // first N-tile column
  const int n1 = n0 + 16;                // second N-tile column

  v8f acc0 = {}, acc1 = {};
  for (int kk = 0; kk < DIM; kk += 4) {
    const int j0 = kk + khalf;
    const float2 af = *(const float2*)&lds[mrow * LDP + j0];   // ds_load_b64
    v2f a; a.x = af.x; a.y = af.y;
    // B[k][n] = M[n][k]  (h = x @ M^T)
    const float2 b0f = *(const float2*)&M[(size_t)n0 * DIM + j0];
    const float2 b1f = *(const float2*)&M[(size_t)n1 * DIM + j0];
    v2f b0; b0.x = b0f.x; b0.y = b0f.y;
    v2f b1; b1.x = b1f.x; b1.y = b1f.y;
    acc0 = wmma_f32_16x16x4(a, b0, acc0);
    acc1 = wmma_f32_16x16x4(a, b1, acc1);
  }
  __syncthreads();   // x-tile no longer needed; reuse LDS for h staging

  // D layout: VGPR v -> row v (lanes 0-15) / row v+8 (lanes 16-31); col = lane%16.
  {
    const int rbase = (lane < 16) ? 0 : 8;
    const int c0 = n0;
#pragma unroll
    for (int v = 0; v < 8; ++v) {
      lds[(rbase + v) * LDP + c0]      = acc0[v];
      lds[(rbase + v) * LDP + c0 + 16] = acc1[v];
    }
  }
  __syncthreads();

  // LayerNorm: 16 threads per row, 16 elements each; butterfly reduce (wave32).
  const int r   = tid >> 4;
  const int seg = tid & 15;
  float s1 = 0.f, s2 = 0.f;
#pragma unroll
  for (int c = 0; c < 16; ++c) {
    const float h = lds[r * LDP + seg * 16 + c];
    s1 += h; s2 += h * h;
  }
#pragma unroll
  for (int m = 1; m < 16; m <<= 1) {
    s1 += __shfl_xor(s1, m, 32);
    s2 += __shfl_xor(s2, m, 32);
  }
  const float mu  = s1 * (1.f / DIM);
  const float inv = rsqrtf(s2 * (1.f / DIM) - mu * mu + 1e-5f);
#pragma unroll
  for (int c = 0; c < 16; ++c) {
    const int col = seg * 16 + c;
    const float h = lds[r * LDP + col];
    xt[(R0 + r) * DIM + col] = (h - mu) * inv * gamma[col] + beta[col];
  }
}

// ---------------------------------------------------------------------------
// Kernel 2: Nk[b,s,i] = sum_j P[i,j]*xt[b,s,j]*cos(2*pi*s/(i*256+j+2)) + x.
// One block per s (2048 blocks). A-tile rows 0..3 = xt[b,s,:], rows 4..15 = 0.
// W_s fragments are synthesized per K-step with v_rcp_f32 + v_cos_f32
// (hw cos computes cos(2*pi*t), exactly our phase convention).
// ---------------------------------------------------------------------------
__global__ __launch_bounds__(256) void k_phi(
    const float* __restrict__ xt, const float* __restrict__ P,
    const float* __restrict__ x, const long long* __restrict__ positions,
    float* __restrict__ out) {
  __shared__ float lds[16 * LDP];

  const int s    = blockIdx.x;
  const int tid  = threadIdx.x;
  const int wave = tid >> 5;
  const int lane = tid & 31;
  const float kpos = (float)positions[s];

  __builtin_prefetch(P, 0, 0);  // global_prefetch_b8

  // Rows 0..3 async from xt (256 threads == 4 rows x 64 float4, one each);
  // rows 4..15 zero-filled with plain LDS stores.
  {
    const int r = tid >> 6, c4 = tid & 63;
    const unsigned loff = (unsigned)(size_t)&lds[r * LDP + c4 * 4];
    async_load_b128(loff, xt + ((size_t)r * SEQ + s) * DIM + c4 * 4);
    const float4 z = make_float4(0.f, 0.f, 0.f, 0.f);
    for (int i = tid; i < 12 * 64; i += 256) {
      const int rr = 4 + (i >> 6), cc4 = i & 63;
      *(float4*)&lds[rr * LDP + cc4 * 4] = z;
    }
  }
  wait_async0();
  __syncthreads();

  const int mrow  = lane & 15;
  const int khalf = (lane >> 4) << 1;
  const int n0 = wave * 32 + mrow;   // output feature i for tile 0
  const int n1 = n0 + 16;            // output feature i for tile 1

  v8f acc0 = {}, acc1 = {};
  for (int kk = 0; kk < DIM; kk += 4) {
    const int j0 = kk + khalf;
    const float2 af = *(const float2*)&lds[mrow * LDP + j0];
    v2f a; a.x = af.x; a.y = af.y;

    const float2 p0 = *(const float2*)&P[(size_t)n0 * DIM + j0];
    const float2 p1 = *(const float2*)&P[(size_t)n1 * DIM + j0];

    // periods[i,j] = i*256 + j + 2 ; t = s/period (revolutions)
    const float base0 = (float)(n0 * DIM + j0 + 2);
    const float base1 = (float)(n1 * DIM + j0 + 2);
    float t00 = kpos * __builtin_amdgcn_rcpf(base0);
    float t01 = kpos * __builtin_amdgcn_rcpf(base0 + 1.f);
    float t10 = kpos * __builtin_amdgcn_rcpf(base1);
    float t11 = kpos * __builtin_amdgcn_rcpf(base1 + 1.f);
    t00 -= floorf(t00);  t01 -= floorf(t01);
    t10 -= floorf(t10);  t11 -= floorf(t11);

    v2f b0, b1;
    b0.x = p0.x * __builtin_amdgcn_cosf(t00);   // v_cos_f32: cos(2*pi*t)
    b0.y = p0.y * __builtin_amdgcn_cosf(t01);
    b1.x = p1.x * __builtin_amdgcn_cosf(t10);
    b1.y = p1.y * __builtin_amdgcn_cosf(t11);

    acc0 = wmma_f32_16x16x4(a, b0, acc0);
    acc1 = wmma_f32_16x16x4(a, b1, acc1);
  }

  // Only rows 0..3 of the D tile are real (b = 0..3), held by lanes 0..15.
  if (lane < 16) {
#pragma unroll
    for (int v = 0; v < BATCH; ++v) {
      const size_t idx0 = ((size_t)v * SEQ + s) * DIM + n0;  // n0 == wave*32+lane
      const size_t idx1 = idx0 + 16;
      out[idx0] = acc0[v] + x[idx0];
      out[idx1] = acc1[v] + x[idx1];
    }
  }
}

// ---------------------------------------------------------------------------
// Host-side launcher. Inputs per setup_inputs(): x, positions, M, P, gamma, beta
// ---------------------------------------------------------------------------
extern "C" void kernel_launch(void* const* d_in, const int* in_sizes, int n_in,
                              void* d_out, int out_size, void* d_ws, size_t ws_size,
                              hipStream_t stream) {
  (void)in_sizes; (void)n_in; (void)out_size; (void)ws_size;
  const float*     x         = (const float*)d_in[0];
  const long long* positions = (const long long*)d_in[1];   // int64 positions
  const float*     M         = (const float*)d_in[2];
  const float*     P         = (const float*)d_in[3];
  const float*     gamma     = (const float*)d_in[4];
  const float*     beta      = (const float*)d_in[5];
  float*           out       = (float*)d_out;
  float*           xt        = (float*)d_ws;   // 8 MB: (B*S) x D fp32

  k_gemm_ln<<<(BATCH * SEQ) / 16, 256, 0, stream>>>(x, M, gamma, beta, xt);
  k_phi<<<SEQ, 256, 0, stream>>>(xt, P, x, positions, out);
}